// PQLinearSuper_68968584839487
// MI455X (gfx1250) — compile-verified
//
#include <hip/hip_runtime.h>
#include <stdint.h>

// PQ-quantized linear: out[M,N] = x[M,K] * W[N,K]^T + bias, W gathered from
// a product-quantization codebook on the fly (never materialized in HBM).
// M=16384 (8*2048), N=4096, K=4096. f32 WMMA 16x16x4 keeps exact precision.
// Data movement: GLOBAL_LOAD_ASYNC_TO_LDS_B128 (ASYNCcnt) + double-buffered
// LDS so chunk c+1 streams from HBM while chunk c runs on the WMMA units.

typedef float v2f __attribute__((ext_vector_type(2)));
typedef float v8f __attribute__((ext_vector_type(8)));

#define TILE_M 128
#define TILE_N 128
#define TILE_K 32
#define LDS_LD 36                    // padded stride (floats): 16B rows, bank-staggered
#define XSZ (TILE_M * LDS_LD)        // floats per X buffer (4608)
#define WSZ (TILE_N * LDS_LD)        // floats per W buffer (4608)
#define NCHUNK (4096 / TILE_K)       // 128 K-chunks

// 16B/lane async copy: copies from global address sbase + voff_byte to LDS at lds_ptr, tracked by
// ASYNCcnt. The LDS destination is passed as a real pointer and ptrtoint'd so
// the shared array escapes into the asm ("memory" clobber then protects it);
// the low 32 bits of a flat shared address are the group-segment byte offset,
// which is exactly the VDST operand of global_load_async_to_lds_*.
__device__ __forceinline__ void async_b128(const float* lds_ptr,
                                           uint32_t voff_byte,
                                           const void* sbase) {
    asm volatile("global_load_async_to_lds_b128 %0, %1, %2"
                 :: "v"((uint32_t)(size_t)lds_ptr), "v"(voff_byte), "s"(sbase)
                 : "memory");
}

__global__ __launch_bounds__(256)
void pq_linear_wmma_f32_async_kernel(const float* __restrict__ x,
                                     const float* __restrict__ centroids,
                                     const float* __restrict__ bias,
                                     const int*   __restrict__ assignments,
                                     float*       __restrict__ out)
{
    // [ Xbuf0 | Xbuf1 | Wbuf0 | Wbuf1 ]  -> 73,728 B of LDS (of 320 KB/WGP)
    __shared__ float smem[2 * XSZ + 2 * WSZ];

    const int n0 = blockIdx.x * TILE_N;     // output-feature tile (N)
    const int m0 = blockIdx.y * TILE_M;     // flattened (b,s) tile (M)

    const int tid  = threadIdx.x;
    const int lane = tid & 31;
    const int wave = tid >> 5;
    const int wm   = wave & 3;              // wave M slot: 4 x 32 rows
    const int wn   = wave >> 2;             // wave N slot: 2 x 64 cols
    const int l15  = lane & 15;
    const int hi   = lane >> 4;             // lane half (0/1)

    // PQ geometry: BX=BY=4, out_dim=in_dim=1024, bs=8, num_blocks=128, NC=256
    const int bxi     = n0 >> 10;
    const int od_base = n0 & 1023;

    // ---- per-thread address precompute -------------------------------------
    // X tile: 1024 float4 transfers/chunk, 4 per thread.
    uint32_t xg[4];
    const float* xlp[4];
#pragma unroll
    for (int i = 0; i < 4; ++i) {
        const int q   = tid + (i << 8);               // 0..1023
        const int row = q >> 3;
        const int cq  = q & 7;
        xg[i]  = (uint32_t)(((m0 + row) * 4096 + (cq << 2)) * 4); // + k*4 at issue
        xlp[i] = &smem[row * LDS_LD + (cq << 2)];                 // buffer 0 dest
    }
    // W tile: 512 codebook gathers/chunk (8 floats each), 2 per thread.
    int olv[2], jj[2];
    const float* wlp[2];
#pragma unroll
    for (int pp = 0; pp < 2; ++pp) {
        const int p = (tid << 1) + pp;                // 0..511
        olv[pp] = p >> 2;                             // local out row 0..127
        jj[pp]  = p & 3;                              // bs=8 block within chunk
        wlp[pp] = &smem[2 * XSZ + olv[pp] * LDS_LD + (jj[pp] << 3)];
    }

    // Issue all async transfers for K-chunk starting at kc into buffer `buf`.
    auto issue_chunk = [&](int kc, int buf) {
        const int bxo = buf * XSZ;                    // float offsets
        const int bwo = buf * WSZ;
        const uint32_t kbyte = (uint32_t)kc * 4u;
#pragma unroll
        for (int i = 0; i < 4; ++i)
            async_b128(xlp[i] + bxo, xg[i] + kbyte, x);

        const int grp = (bxi << 2) + (kc >> 10);      // (bx,by) codebook group
        const int* __restrict__ asg = assignments + (size_t)grp * (128 * 1024);
        const int nb_base = (kc & 1023) >> 3;
#pragma unroll
        for (int pp = 0; pp < 2; ++pp) {
            const int a = asg[(nb_base + jj[pp]) * 1024 + od_base + olv[pp]];
            const uint32_t voff = (uint32_t)(grp * 8192 + (a << 5)); // bytes
            async_b128(wlp[pp] + bwo,     voff,      centroids);
            async_b128(wlp[pp] + bwo + 4, voff + 16, centroids);
        }
    };

    const v8f vzero = {0.f,0.f,0.f,0.f,0.f,0.f,0.f,0.f};
    v8f acc[2][4];
#pragma unroll
    for (int mi = 0; mi < 2; ++mi)
#pragma unroll
        for (int ni = 0; ni < 4; ++ni)
            acc[mi][ni] = vzero;

    // Prologue: start chunk 0 into buffer 0.
    issue_chunk(0, 0);

    for (int c = 0; c < NCHUNK; ++c) {
        const int cur = c & 1;

        if (c + 1 < NCHUNK) {
            issue_chunk((c + 1) * TILE_K, cur ^ 1);   // stream next chunk
            // 8 async ops outstanding for next chunk; oldest 8 (= cur) done:
            asm volatile("s_wait_asynccnt 0x8" ::: "memory");
        } else {
            asm volatile("s_wait_asynccnt 0x0" ::: "memory");
        }
        __syncthreads();   // cur tile visible to all waves

        const float* Xb = smem + cur * XSZ;
        const float* Wb = smem + 2 * XSZ + cur * WSZ;

        // ---- 8 K-steps x (2 M-subtiles x 4 N-subtiles) WMMAs ----
#pragma unroll
        for (int kk = 0; kk < TILE_K; kk += 4) {
            const int kc2 = kk + (hi << 1);  // A/B VGPR0 holds K=kc2, VGPR1 kc2+1
            const v2f a0 = *(const v2f*)&Xb[(wm * 32 +      l15) * LDS_LD + kc2];
            const v2f a1 = *(const v2f*)&Xb[(wm * 32 + 16 + l15) * LDS_LD + kc2];
#pragma unroll
            for (int ni = 0; ni < 4; ++ni) {
                const v2f b = *(const v2f*)&Wb[(wn * 64 + ni * 16 + l15) * LDS_LD + kc2];
                acc[0][ni] = __builtin_amdgcn_wmma_f32_16x16x4_f32(
                    false, a0, false, b, (short)0, acc[0][ni], false, false);
                acc[1][ni] = __builtin_amdgcn_wmma_f32_16x16x4_f32(
                    false, a1, false, b, (short)0, acc[1][ni], false, false);
            }
        }

        __syncthreads();   // all waves done reading cur before it is re-filled
    }

    // ---- epilogue: C/D layout (VGPR v -> rows v / v+8 by lane half) ----
#pragma unroll
    for (int mi = 0; mi < 2; ++mi) {
        const int r0 = m0 + wm * 32 + mi * 16 + (hi << 3);
#pragma unroll
        for (int ni = 0; ni < 4; ++ni) {
            const int col = n0 + wn * 64 + ni * 16 + l15;
            const float bv = bias[col];
#pragma unroll
            for (int v = 0; v < 8; ++v) {
                out[(size_t)(r0 + v) * 4096 + col] = acc[mi][ni][v] + bv;
            }
        }
    }
}

extern "C" void kernel_launch(void* const* d_in, const int* in_sizes, int n_in,
                              void* d_out, int out_size, void* d_ws, size_t ws_size,
                              hipStream_t stream) {
    (void)in_sizes; (void)n_in; (void)d_ws; (void)ws_size; (void)out_size;
    const float* x           = (const float*)d_in[0];
    const float* centroids   = (const float*)d_in[1];
    const float* bias        = (const float*)d_in[2];
    const int*   assignments = (const int*)d_in[3];
    float*       out         = (float*)d_out;

    // Grid: N tiles (32) x M tiles (128); 256 threads = 8 wave32s per block.
    dim3 grid(4096 / TILE_N, 16384 / TILE_M);
    dim3 block(256);
    pq_linear_wmma_f32_async_kernel<<<grid, block, 0, stream>>>(
        x, centroids, bias, assignments, out);
}